// TT_kernel_component_14448269984287
// MI455X (gfx1250) — compile-verified
//
#include <hip/hip_runtime.h>
#include <math.h>

// ---------------------------------------------------------------------------
// Problem constants (from reference): n=8192, d_feat=32, r1=r2=16, B=32768.
// out[b,a,c] = sum_n K[idx[b], n] * TT[a, n, c],  K = exp(-0.5*d2/ls^2),
// ls = softplus(median of positive pairwise distances of side_data).
// ---------------------------------------------------------------------------
#define N_PTS   8192
#define D_FEAT  32
#define NCOLS   256          // r1*r2
#define NBINS   8192
#define D2_MAX  256.0f       // d2 ~ 2*chi2(32): mean 64, std 16 -> [0,256) safe

typedef __attribute__((ext_vector_type(2))) float v2f;
typedef __attribute__((ext_vector_type(8))) float v8f;

// ---------------------------------------------------------------------------
// 1) Row squared norms of X [8192,32]
// ---------------------------------------------------------------------------
__global__ void rownorm_kernel(const float* __restrict__ X, float* __restrict__ sq) {
  int i = blockIdx.x * blockDim.x + threadIdx.x;
  if (i < N_PTS) {
    const float4* x4 = (const float4*)(X + i * D_FEAT);
    float s = 0.f;
#pragma unroll
    for (int k = 0; k < D_FEAT / 4; ++k) {
      float4 v = x4[k];
      s += v.x * v.x + v.y * v.y + v.z * v.z + v.w * v.w;
    }
    sq[i] = s;
  }
}

// ---------------------------------------------------------------------------
// 2) Histogram of strictly-positive pairwise squared distances.
//    Grid (128,128): each block does a 64x64 tile with LDS-staged X rows.
// ---------------------------------------------------------------------------
__global__ void hist_kernel(const float* __restrict__ X, unsigned* __restrict__ hist) {
  __shared__ float Xi[64 * D_FEAT];       //  8 KB
  __shared__ float Xj[64 * D_FEAT];       //  8 KB
  __shared__ unsigned h[NBINS];           // 32 KB
  const int tid = threadIdx.x;            // 256 threads
  for (int b = tid; b < NBINS; b += 256) h[b] = 0u;
  const int i0 = blockIdx.y * 64, j0 = blockIdx.x * 64;
  for (int t = tid; t < 64 * D_FEAT; t += 256) {
    Xi[t] = X[i0 * D_FEAT + t];
    Xj[t] = X[j0 * D_FEAT + t];
  }
  __syncthreads();
  for (int t = tid; t < 64 * 64; t += 256) {
    const int ii = t >> 6, jj = t & 63;
    float d2 = 0.f;
#pragma unroll
    for (int k = 0; k < D_FEAT; ++k) {
      float df = Xi[ii * D_FEAT + k] - Xj[jj * D_FEAT + k];
      d2 += df * df;
    }
    if (d2 > 0.f) {                       // strictly positive distances only
      int b = (int)(d2 * ((float)NBINS / D2_MAX));
      if (b > NBINS - 1) b = NBINS - 1;
      atomicAdd(&h[b], 1u);
    }
  }
  __syncthreads();
  for (int b = tid; b < NBINS; b += 256)
    if (h[b]) atomicAdd(&hist[b], h[b]);
}

// ---------------------------------------------------------------------------
// 3) Median selection from histogram -> ls = softplus(med) -> scale = -0.5/ls^2
// ---------------------------------------------------------------------------
__global__ void median_kernel(const unsigned* __restrict__ hist, float* __restrict__ scl) {
  __shared__ unsigned long long csum[257];
  const int tid = threadIdx.x;            // 256 threads
  unsigned long long s = 0;
  for (int k = 0; k < NBINS / 256; ++k) s += hist[tid * (NBINS / 256) + k];
  csum[tid + 1] = s;
  if (tid == 0) csum[0] = 0;
  __syncthreads();
  if (tid == 0) {
    for (int i = 1; i <= 256; ++i) csum[i] += csum[i - 1];
    unsigned long long total = csum[256];
    float med;
    if (total == 0) {
      med = 1.0f;
    } else {
      double target = 0.5 * (double)(total - 1);            // fractional rank
      unsigned long long r = (unsigned long long)target;
      int c = 0;
      while (c < 255 && csum[c + 1] <= r) ++c;
      unsigned long long acc = csum[c];
      int b = c * (NBINS / 256), bend = b + (NBINS / 256);
      unsigned hb = 0;
      while (b < bend) { hb = hist[b]; if (acc + hb > r) break; acc += hb; ++b; }
      if (b == bend) { b = bend - 1; hb = hist[b]; }
      double frac = hb ? ((target - (double)acc) + 0.5) / (double)hb : 0.5;
      double d2v = ((double)b + frac) * (double)(D2_MAX / (float)NBINS);
      med = sqrtf((float)(d2v > 0.0 ? d2v : 0.0));
    }
    // ls_factor = 1.0 ; softplus
    float ls = (med > 20.f) ? med : log1pf(expf(med));
    scl[0] = -0.5f / (ls * ls);
  }
}

// ---------------------------------------------------------------------------
// 4) Fused kernel: Tp[m, j] = sum_n exp(scale*d2(m,n)) * V[n, j]
//    V[n, a*16+c] = TT[a*131072 + n*16 + c].
//    Block: 256 threads (8 waves), M_BLK = 64 rows, full 256 cols, loop n in
//    chunks of 32. Both GEMMs use V_WMMA_F32_16X16X4_F32; the K-tile is routed
//    through LDS to convert D-layout -> A-layout.
// ---------------------------------------------------------------------------
#define PITCH 34   // even -> 8B-aligned ds b64 reads; odd-ish stride -> no bank conflicts

__global__ void __launch_bounds__(256)
fused_gemm_kernel(const float* __restrict__ X, const float* __restrict__ TT,
                  const float* __restrict__ sq, const float* __restrict__ scl,
                  float* __restrict__ Tp) {
  __shared__ float Xm[64 * D_FEAT];       //  8 KB, A rows for Gram
  __shared__ float Xn[32 * D_FEAT];       //  4 KB, B rows for Gram (chunk)
  __shared__ float Vl[NCOLS * PITCH];     // ~34 KB, V chunk transposed [j][n_local]
  __shared__ float Kl[64 * PITCH];        // ~8.5 KB, exp'd Gram tile (A of 2nd GEMM)
  __shared__ float sqm[64];
  __shared__ float sqn[32];

  const int tid  = threadIdx.x;
  const int wave = tid >> 5;
  const int lane = tid & 31;
  const int lr   = lane & 15;             // M (A) / N (B,C/D) index within tile
  const int lh   = lane >> 4;             // lane half -> K offset pair
  const int m0   = blockIdx.x * 64;
  const float scale = scl[0];

  // stage the block's X rows + norms once
  for (int t = tid; t < 64 * D_FEAT; t += 256) Xm[t] = X[m0 * D_FEAT + t];
  if (tid < 64) sqm[tid] = sq[m0 + tid];

  v8f acc[8];
#pragma unroll
  for (int jt = 0; jt < 8; ++jt)
#pragma unroll
    for (int v = 0; v < 8; ++v) acc[jt][v] = 0.f;

  const int mt = wave & 3;                // Gram tile row / 2nd-GEMM tile row
  const int nt = wave >> 2;               // Gram tile col (chunk has 2 n-tiles)
  const int jh = wave >> 2;               // j-half for 2nd GEMM

  for (int nc = 0; nc < N_PTS / 32; ++nc) {
    const int n0 = nc * 32;
    // ---- stage chunk: X_n rows, sq_n, V chunk transposed --------------------
    for (int t = tid; t < 32 * D_FEAT; t += 256) Xn[t] = X[n0 * D_FEAT + t];
    if (tid < 32) sqn[tid] = sq[n0 + tid];
    {
      const int j = tid;                  // 0..255 ; a=j>>4, c=j&15
      const float* src = TT + (size_t)(j >> 4) * (N_PTS * 16) + (size_t)n0 * 16 + (j & 15);
#pragma unroll 8
      for (int nl = 0; nl < 32; ++nl) Vl[j * PITCH + nl] = src[nl * 16];
    }
    __syncthreads();

    // ---- Gram tile: G = X_m(16x32) @ X_n(16x32)^T via 8x WMMA f32 16x16x4 ---
    v8f g;
#pragma unroll
    for (int v = 0; v < 8; ++v) g[v] = 0.f;
#pragma unroll
    for (int s = 0; s < 8; ++s) {
      const int kb = 4 * s + (lh << 1);
      v2f av = *(const v2f*)&Xm[(mt * 16 + lr) * D_FEAT + kb];
      v2f bv = *(const v2f*)&Xn[(nt * 16 + lr) * D_FEAT + kb];
      g = __builtin_amdgcn_wmma_f32_16x16x4_f32(false, av, false, bv,
                                                (short)0, g, false, false);
    }
    // ---- K tile = exp(scale * max(d2,0)) -> LDS in A-consumable layout ------
    {
      const float sn = sqn[nt * 16 + lr];
#pragma unroll
      for (int v = 0; v < 8; ++v) {
        const int rloc = mt * 16 + v + lh * 8;      // C/D layout: M = v (+8 for hi half)
        float d2 = sqm[rloc] + sn - 2.f * g[v];
        d2 = fmaxf(d2, 0.f);
        Kl[rloc * PITCH + nt * 16 + lr] = __expf(scale * d2);
      }
    }
    __syncthreads();

    // ---- 2nd GEMM: acc[mt2, jt] += K_tile(16x32) @ V_chunk(32x16) -----------
    v2f afr[8];
#pragma unroll
    for (int s = 0; s < 8; ++s) {
      const int kb = 4 * s + (lh << 1);
      afr[s] = *(const v2f*)&Kl[(mt * 16 + lr) * PITCH + kb];
    }
#pragma unroll
    for (int jt = 0; jt < 8; ++jt) {
      const int j0 = (jh * 8 + jt) * 16;
      v8f a = acc[jt];
#pragma unroll
      for (int s = 0; s < 8; ++s) {
        const int kb = 4 * s + (lh << 1);
        v2f bv = *(const v2f*)&Vl[(j0 + lr) * PITCH + kb];
        a = __builtin_amdgcn_wmma_f32_16x16x4_f32(false, afr[s], false, bv,
                                                  (short)0, a, false, false);
      }
      acc[jt] = a;
    }
    __syncthreads();   // protect Xn/Vl/Kl before next chunk's staging
  }

  // ---- write Tp[m, j] (coalesced across lanes) ------------------------------
#pragma unroll
  for (int jt = 0; jt < 8; ++jt) {
    const int j0 = (jh * 8 + jt) * 16;
#pragma unroll
    for (int v = 0; v < 8; ++v) {
      const int rloc = mt * 16 + v + lh * 8;
      Tp[(size_t)(m0 + rloc) * NCOLS + j0 + lr] = acc[jt][v];
    }
  }
}

// ---------------------------------------------------------------------------
// 5) Gather: out[b, :] = Tp[idx[b], :]   (256 floats = 64 float4 per row)
// ---------------------------------------------------------------------------
__global__ void gather_kernel(const float* __restrict__ Tp, const int* __restrict__ idx,
                              float* __restrict__ out) {
  const int b = blockIdx.x;
  const int row = idx[b];
  const float4* src = (const float4*)(Tp + (size_t)row * NCOLS);
  float4* dst = (float4*)(out + (size_t)b * NCOLS);
  dst[threadIdx.x] = src[threadIdx.x];    // 64 threads x 16B
}

// ---------------------------------------------------------------------------
// Workspace layout (needs ~8.2 MB):
//   [0        , 32768)  : sq[8192]            f32
//   [32768    , 65536)  : hist[8192]          u32
//   [65536    , 65540)  : scale               f32
//   [131072   , +8 MB)  : Tp[8192][256]       f32
// ---------------------------------------------------------------------------
extern "C" void kernel_launch(void* const* d_in, const int* in_sizes, int n_in,
                              void* d_out, int out_size, void* d_ws, size_t ws_size,
                              hipStream_t stream) {
  (void)n_in; (void)out_size; (void)ws_size;
  const float* TT  = (const float*)d_in[0];   // [16, 8192, 16]
  const float* X   = (const float*)d_in[1];   // [8192, 32]
  const int*   idx = (const int*)d_in[2];     // [B]
  float* out = (float*)d_out;                 // [B, 16, 16]
  const int B = in_sizes[2];

  char* ws = (char*)d_ws;
  float*    sq   = (float*)(ws + 0);
  unsigned* hist = (unsigned*)(ws + 32768);
  float*    scl  = (float*)(ws + 65536);
  float*    Tp   = (float*)(ws + 131072);

  hipMemsetAsync(hist, 0, NBINS * sizeof(unsigned), stream);
  rownorm_kernel<<<N_PTS / 256, 256, 0, stream>>>(X, sq);
  hist_kernel<<<dim3(N_PTS / 64, N_PTS / 64), 256, 0, stream>>>(X, hist);
  median_kernel<<<1, 256, 0, stream>>>(hist, scl);
  fused_gemm_kernel<<<N_PTS / 64, 256, 0, stream>>>(X, TT, sq, scl, Tp);
  gather_kernel<<<B, 64, 0, stream>>>(Tp, idx, out);
}